// DEQIPFixedPoint_18193481466202
// MI455X (gfx1250) — compile-verified
//
#include <hip/hip_runtime.h>
#include <math.h>

typedef __attribute__((ext_vector_type(2))) float v2f;
typedef __attribute__((ext_vector_type(8))) float v8f;

#define HH 128
#define WWI 128
#define NB 8
#define HW 16384          // 128*128
#define NPIX 131072       // 8*16384
#define CHN 64
#define MEMM 5
#define LAMR 1e-4f
#define TOLR 0.01f
#define ETAR 0.5f
#define MAXIT 30

struct DeqState { int done; int cur_k; };

__device__ __forceinline__ float relu(float v) { return v > 0.f ? v : 0.f; }

// wait for this wave's async global->LDS transfers
__device__ __forceinline__ void wait_async0() {
#if __has_builtin(__builtin_amdgcn_s_wait_asynccnt)
    __builtin_amdgcn_s_wait_asynccnt(0);
#else
    asm volatile("s_wait_asynccnt 0x0" ::: "memory");
#endif
}

// ------------------------------------------------------------ blur pipeline
// Block = 2 output rows of one image; LDS-stage a 6x134 halo tile.
// R = conv5(X) - y     (XLA conv = cross-correlation, SAME pad 2)
__global__ void k_blurA(const float* __restrict__ in, int in_bstr,
                        const float* __restrict__ hist, const DeqState* __restrict__ sel,
                        const float* __restrict__ y, const float* __restrict__ A0,
                        float* __restrict__ R, const DeqState* __restrict__ gate)
{
    __shared__ float xt[6 * 134];
    if (gate && gate->done) return;
    int blk = blockIdx.x;                 // 0..511
    int b = blk >> 6, h0 = (blk & 63) * 2;
    const float* src = hist ? (hist + (size_t)(b * MEMM + (sel->cur_k % MEMM)) * HW)
                            : (in + (size_t)b * in_bstr);
    for (int i = threadIdx.x; i < 6 * 134; i += 256) {
        int r = i / 134, c = i % 134;
        int hh = h0 + r - 2, ww = c - 2;
        xt[i] = (hh >= 0 && hh < HH && ww >= 0 && ww < WWI) ? src[hh * WWI + ww] : 0.f;
    }
    __syncthreads();
    int lr = threadIdx.x >> 7, w = threadIdx.x & 127;
    int h = h0 + lr;
    float acc = 0.f;
#pragma unroll
    for (int r = 0; r < 5; r++)
#pragma unroll
        for (int s = 0; s < 5; s++)
            acc += xt[(lr + r) * 134 + (w + s)] * A0[r * 5 + s];
    int q = b * HW + h * WWI + w;
    R[q] = acc - y[q];
}

// Xg = X - ETA * conv5(R, flip(A0));  stages R, reads X center directly
__global__ void k_blurB(const float* __restrict__ in, int in_bstr,
                        const float* __restrict__ hist, const DeqState* __restrict__ sel,
                        const float* __restrict__ R, const float* __restrict__ A0,
                        float* __restrict__ Xg, const DeqState* __restrict__ gate)
{
    __shared__ float rt[6 * 134];
    if (gate && gate->done) return;
    int blk = blockIdx.x;
    int b = blk >> 6, h0 = (blk & 63) * 2;
    const float* src = hist ? (hist + (size_t)(b * MEMM + (sel->cur_k % MEMM)) * HW)
                            : (in + (size_t)b * in_bstr);
    const float* Rb = R + (size_t)b * HW;
    for (int i = threadIdx.x; i < 6 * 134; i += 256) {
        int r = i / 134, c = i % 134;
        int hh = h0 + r - 2, ww = c - 2;
        rt[i] = (hh >= 0 && hh < HH && ww >= 0 && ww < WWI) ? Rb[hh * WWI + ww] : 0.f;
    }
    __syncthreads();
    int lr = threadIdx.x >> 7, w = threadIdx.x & 127;
    int h = h0 + lr;
    float acc = 0.f;
#pragma unroll
    for (int r = 0; r < 5; r++)
#pragma unroll
        for (int s = 0; s < 5; s++)
            acc += rt[(lr + r) * 134 + (w + s)] * A0[(4 - r) * 5 + (4 - s)];
    Xg[b * HW + h * WWI + w] = src[h * WWI + w] - ETAR * acc;
}

// ------------------------------------------------- conv1: 1->64ch 3x3, WMMA
// Block = one image row (8 waves x 16 px). Stage 4x132 Xg halo (row 3 = zeros
// so padded taps t=9..11 need no guard) + W1 zero-padded to [64][12] in LDS.
// GEMM per wave: M=16 px, N=64 (4 tiles), K=12 (3 k-steps) -> 12 v_wmma.
__global__ void k_conv1(const float* __restrict__ Xg, const float* __restrict__ W1,
                        const float* __restrict__ b1, float* __restrict__ Z,
                        const DeqState* __restrict__ gate)
{
    __shared__ float xt[4 * 132];
    __shared__ float w1l[64 * 12];
    if (gate && gate->done) return;
    int row = blockIdx.x;                 // 0..1023 = b*128 + h
    int b = row >> 7, h = row & 127;
    const float* src = Xg + (size_t)b * HW;
    for (int i = threadIdx.x; i < 4 * 132; i += 256) {
        int r = i / 132, c = i % 132;
        int hh = h + r - 1, ww = c - 1;
        xt[i] = (r < 3 && hh >= 0 && hh < HH && ww >= 0 && ww < WWI) ? src[hh * WWI + ww] : 0.f;
    }
    for (int i = threadIdx.x; i < 64 * 12; i += 256) {
        int c = i / 12, t = i % 12;
        w1l[i] = (t <= 8) ? W1[c * 9 + t] : 0.f;
    }
    __syncthreads();

    int wv = threadIdx.x >> 5, lane = threadIdx.x & 31;
    int m = lane & 15, hi = lane >> 4;
    int w0 = wv * 16, wm = w0 + m;
    int base = row * WWI + w0;

    v8f acc[4] = {};
#pragma unroll
    for (int kk = 0; kk < 3; kk++) {
        int t0 = kk * 4 + 2 * hi, t1 = t0 + 1;
        int r0 = (t0 * 11) >> 5, d0 = t0 - 3 * r0;   // t/3, t%3 for t<12
        int r1 = (t1 * 11) >> 5, d1 = t1 - 3 * r1;
        v2f a; a.x = xt[r0 * 132 + wm + d0]; a.y = xt[r1 * 132 + wm + d1];
#pragma unroll
        for (int j = 0; j < 4; j++) {
            int c = j * 16 + m;
            v2f bb; bb.x = w1l[c * 12 + t0]; bb.y = w1l[c * 12 + t1];
            acc[j] = __builtin_amdgcn_wmma_f32_16x16x4_f32(
                false, a, false, bb, (short)0, acc[j], false, false);
        }
    }
#pragma unroll
    for (int j = 0; j < 4; j++) {
        int c = j * 16 + m;
        float bias = b1[c];
#pragma unroll
        for (int r = 0; r < 8; r++) {
            int p = base + r + 8 * hi;
            Z[(size_t)p * CHN + c] = relu(acc[j][r] + bias);
        }
    }
}

// ------------------------------------------------ conv2 stage 1: WMMA GEMM
// T[p][t] = sum_c Z[p][c] * W2[c][t]   (t=0..8 padded to 16)
// Block = one row (128 px). Z tile (128x64, 32KB) staged via
// global_load_async_to_lds_b128 (coalesced, ASYNCcnt) into a 68-float-pitch
// LDS tile (bank-conflict free for the stride-68 A reads). W2 zero-padded
// to [64][16] in LDS. GEMM per wave: M=16 px, N=16 taps, K=64 -> 16 v_wmma.
__global__ void k_conv2(const float* __restrict__ Z, const float* __restrict__ W2,
                        float* __restrict__ T, const DeqState* __restrict__ gate)
{
    __shared__ float zt[128 * 68];
    __shared__ float w2l[64 * 16];
    if (gate && gate->done) return;
    int row = blockIdx.x;                 // 128-pixel group
    int pbase = row * 128;
    const float* gsrc = Z + (size_t)pbase * CHN;
    // 2048 x 16B chunks: chunk ci -> pixel ci>>4, quarter ci&15
#pragma unroll
    for (int it = 0; it < 8; it++) {
        int ci = it * 256 + threadIdx.x;
        int pl = ci >> 4, part = ci & 15;
        unsigned long long ga = (unsigned long long)(uintptr_t)(gsrc + ci * 4);
        unsigned la = (unsigned)(uintptr_t)&zt[pl * 68 + part * 4];
        asm volatile("global_load_async_to_lds_b128 %0, %1, off"
                     :: "v"(la), "v"(ga) : "memory");
    }
    for (int i = threadIdx.x; i < 64 * 16; i += 256) {
        int c = i >> 4, t = i & 15;
        w2l[i] = (t < 9) ? W2[c * 9 + t] : 0.f;
    }
    wait_async0();
    __syncthreads();

    int wv = threadIdx.x >> 5, lane = threadIdx.x & 31;
    int m = lane & 15, hi = lane >> 4;
    int base = pbase + wv * 16;
    const float* zl = &zt[(wv * 16 + m) * 68];

    v8f acc = {};
#pragma unroll
    for (int kk = 0; kk < 16; kk++) {
        int c0 = kk * 4 + 2 * hi;
        v2f a; a.x = zl[c0]; a.y = zl[c0 + 1];
        v2f bb; bb.x = w2l[c0 * 16 + m]; bb.y = w2l[(c0 + 1) * 16 + m];
        acc = __builtin_amdgcn_wmma_f32_16x16x4_f32(
            false, a, false, bb, (short)0, acc, false, false);
    }
#pragma unroll
    for (int r = 0; r < 8; r++) {
        int p = base + r + 8 * hi;
        T[(size_t)p * 16 + m] = acc[r];
    }
}

// ------------------------------ conv2 stage 2 + residual partial reduction
// F[q] = Xg[q] + 0.1*(b2 + sum_t T[q+off(t)][t]);  optional ||F-xn||^2, ||F||^2
__global__ void k_finish(const float* __restrict__ T, const float* __restrict__ Xg,
                         const float* __restrict__ b2,
                         float* __restrict__ out, int out_bstr,
                         const float* __restrict__ xn,
                         float* __restrict__ pA, float* __restrict__ pB,
                         const DeqState* __restrict__ gate)
{
    __shared__ float sA[256], sB[256];
    if (gate && gate->done) return;
    int q = blockIdx.x * blockDim.x + threadIdx.x;
    int b = q >> 14, hw = q & 16383;
    int h = hw >> 7, w = hw & 127;
    float s = b2[0];
#pragma unroll
    for (int t = 0; t < 9; t++) {
        int dy = t / 3 - 1, dx = t % 3 - 1;
        int hh = h + dy, ww = w + dx;
        if (hh >= 0 && hh < HH && ww >= 0 && ww < WWI)
            s += T[((size_t)(b * HW + hh * WWI + ww)) * 16 + t];
    }
    float fv = Xg[q] + 0.1f * s;
    out[(size_t)b * out_bstr + hw] = fv;

    if (pA) {
        float d = fv - xn[q];
        int tid = threadIdx.x;
        sA[tid] = d * d; sB[tid] = fv * fv;
        __syncthreads();
        for (int st = 128; st > 0; st >>= 1) {
            if (tid < st) { sA[tid] += sA[tid + st]; sB[tid] += sB[tid + st]; }
            __syncthreads();
        }
        if (tid == 0) { pA[blockIdx.x] = sA[0]; pB[blockIdx.x] = sB[0]; }
    }
}

// ----------------------------------------------------------- Anderson pieces
__global__ void k_gram(const float* __restrict__ Xh, const float* __restrict__ Fh,
                       float* __restrict__ Hs, const DeqState* __restrict__ gate)
{
    __shared__ float sm[256];
    if (gate->done) return;
    int b = blockIdx.x, i = blockIdx.y, j = blockIdx.z;
    const float* Xi = Xh + (size_t)(b * MEMM + i) * HW;
    const float* Fi = Fh + (size_t)(b * MEMM + i) * HW;
    const float* Xj = Xh + (size_t)(b * MEMM + j) * HW;
    const float* Fj = Fh + (size_t)(b * MEMM + j) * HW;
    float s = 0.f;
    for (int f = threadIdx.x; f < HW; f += 256)
        s += (Fi[f] - Xi[f]) * (Fj[f] - Xj[f]);
    int tid = threadIdx.x;
    sm[tid] = s; __syncthreads();
    for (int st = 128; st > 0; st >>= 1) {
        if (tid < st) sm[tid] += sm[tid + st];
        __syncthreads();
    }
    if (tid == 0) Hs[b * 25 + i * 5 + j] = sm[0];
}

__global__ void k_solve(const float* __restrict__ Hs, float* __restrict__ alpha,
                        int n, const DeqState* __restrict__ gate)
{
    if (gate->done) return;
    int b = threadIdx.x;
    if (b >= NB) return;
    float Am[6][6], rhs[6], sol[6];
    int N1 = n + 1;
    Am[0][0] = 0.f; rhs[0] = 1.f;
    for (int j = 1; j <= n; j++) Am[0][j] = 1.f;
    for (int i = 1; i <= n; i++) {
        Am[i][0] = 1.f; rhs[i] = 0.f;
        for (int j = 1; j <= n; j++)
            Am[i][j] = Hs[b * 25 + (i - 1) * 5 + (j - 1)] + (i == j ? LAMR : 0.f);
    }
    for (int col = 0; col < N1; col++) {
        int piv = col; float mx = fabsf(Am[col][col]);
        for (int r = col + 1; r < N1; r++)
            if (fabsf(Am[r][col]) > mx) { mx = fabsf(Am[r][col]); piv = r; }
        if (piv != col) {
            for (int c = 0; c < N1; c++) { float t = Am[col][c]; Am[col][c] = Am[piv][c]; Am[piv][c] = t; }
            float t = rhs[col]; rhs[col] = rhs[piv]; rhs[piv] = t;
        }
        float inv = 1.f / Am[col][col];
        for (int r = col + 1; r < N1; r++) {
            float f = Am[r][col] * inv;
            for (int c = col; c < N1; c++) Am[r][c] -= f * Am[col][c];
            rhs[r] -= f * rhs[col];
        }
    }
    for (int i = N1 - 1; i >= 0; i--) {
        float s = rhs[i];
        for (int c = i + 1; c < N1; c++) s -= Am[i][c] * sol[c];
        sol[i] = s / Am[i][i];
    }
    for (int i = 1; i <= n; i++) alpha[b * MEMM + (i - 1)] = sol[i];
}

__global__ void k_combine(const float* __restrict__ Fh, const float* __restrict__ alpha,
                          int n, int slot, float* __restrict__ xn,
                          float* __restrict__ Xh, const DeqState* __restrict__ gate)
{
    if (gate->done) return;
    int q = blockIdx.x * blockDim.x + threadIdx.x;
    int b = q >> 14, hw = q & 16383;
    float s = 0.f;
    for (int i = 0; i < n; i++)
        s += alpha[b * MEMM + i] * Fh[(size_t)(b * MEMM + i) * HW + hw];
    xn[q] = s;
    Xh[(size_t)(b * MEMM + slot) * HW + hw] = s;
}

__global__ void k_resupdate(const float* __restrict__ pA, const float* __restrict__ pB,
                            int nblocks, int k, DeqState* st)
{
    __shared__ float sA[256], sB[256];
    if (st->done) return;
    float a = 0.f, bb = 0.f;
    for (int i = threadIdx.x; i < nblocks; i += 256) { a += pA[i]; bb += pB[i]; }
    int tid = threadIdx.x;
    sA[tid] = a; sB[tid] = bb; __syncthreads();
    for (int s = 128; s > 0; s >>= 1) {
        if (tid < s) { sA[tid] += sA[tid + s]; sB[tid] += sB[tid + s]; }
        __syncthreads();
    }
    if (tid == 0) {
        float res = sqrtf(sA[0]) / (1e-5f + sqrtf(sB[0]));
        st->cur_k = k;
        if (res < TOLR) st->done = 1;
    }
}

__global__ void k_copy(float* __restrict__ dst, int dstb,
                       const float* __restrict__ src, int srcb)
{
    int q = blockIdx.x * blockDim.x + threadIdx.x;
    int b = q >> 14, hw = q & 16383;
    dst[(size_t)b * dstb + hw] = src[(size_t)b * srcb + hw];
}

__global__ void k_init(DeqState* st)
{
    if (threadIdx.x == 0) { st->done = 0; st->cur_k = 1; }
}

// ----------------------------------------------------------------- launcher
extern "C" void kernel_launch(void* const* d_in, const int* in_sizes, int n_in,
                              void* d_out, int out_size, void* d_ws, size_t ws_size,
                              hipStream_t stream)
{
    const float* y  = (const float*)d_in[0];
    const float* X0 = (const float*)d_in[1];
    // d_in[2] = Z0 (unused by the reference)
    const float* A0 = (const float*)d_in[3];
    const float* W1 = (const float*)d_in[4];
    const float* b1 = (const float*)d_in[5];
    const float* W2 = (const float*)d_in[6];
    const float* b2 = (const float*)d_in[7];
    float* out = (float*)d_out;

    float* ws = (float*)d_ws;
    size_t o = 0;
    float* Xg = ws + o; o += NPIX;
    float* R  = ws + o; o += NPIX;
    float* Z  = ws + o; o += (size_t)NPIX * CHN;
    float* T  = ws + o; o += (size_t)NPIX * 16;
    float* Xh = ws + o; o += (size_t)NB * MEMM * HW;
    float* Fh = ws + o; o += (size_t)NB * MEMM * HW;
    float* xn = ws + o; o += NPIX;
    float* Hs = ws + o; o += 256;
    float* al = ws + o; o += 64;
    float* pA = ws + o; o += 512;
    float* pB = ws + o; o += 512;
    DeqState* st = (DeqState*)(ws + o);

    const int TB = 256;
    const int GP = NPIX / TB;          // pixel-parallel grid (512)
    const int GR = NPIX / 128;         // row-based WMMA grid (1024)

    auto fapply = [&](const float* in, int in_bstr, const float* hist_sel,
                      float* outp, int out_bstr, bool resid, const DeqState* gate) {
        k_blurA<<<GP, TB, 0, stream>>>(in, in_bstr, hist_sel, st, y, A0, R, gate);
        k_blurB<<<GP, TB, 0, stream>>>(in, in_bstr, hist_sel, st, R, A0, Xg, gate);
        k_conv1<<<GR, TB, 0, stream>>>(Xg, W1, b1, Z, gate);
        k_conv2<<<GR, TB, 0, stream>>>(Z, W2, T, gate);
        k_finish<<<GP, TB, 0, stream>>>(T, Xg, b2, outp, out_bstr,
                                        resid ? xn : nullptr,
                                        resid ? pA : nullptr,
                                        resid ? pB : nullptr, gate);
    };

    k_init<<<1, 32, 0, stream>>>(st);
    // X[:,0] = X0
    k_copy<<<GP, TB, 0, stream>>>(Xh, MEMM * HW, X0, HW);
    // F[:,0] = f(X0)
    fapply(X0, HW, nullptr, Fh, MEMM * HW, false, nullptr);
    // X[:,1] = F[:,0]
    k_copy<<<GP, TB, 0, stream>>>(Xh + HW, MEMM * HW, Fh, MEMM * HW);
    // F[:,1] = f(F[:,0])
    fapply(Fh, MEMM * HW, nullptr, Fh + HW, MEMM * HW, false, nullptr);

    for (int k = 2; k < MAXIT; k++) {
        int n = k < MEMM ? k : MEMM;
        int slot = k % MEMM;
        k_gram<<<dim3(NB, n, n), TB, 0, stream>>>(Xh, Fh, Hs, st);
        k_solve<<<1, NB, 0, stream>>>(Hs, al, n, st);
        k_combine<<<GP, TB, 0, stream>>>(Fh, al, n, slot, xn, Xh, st);
        fapply(xn, HW, nullptr, Fh + (size_t)slot * HW, MEMM * HW, true, st);
        k_resupdate<<<1, TB, 0, stream>>>(pA, pB, GP, k, st);
    }

    // out = f(X[:, cur_k % M])  -- slot selected on-device via state
    fapply(nullptr, 0, Xh, out, HW, false, nullptr);
}